// MultiHeadAttention_44461501448865
// MI455X (gfx1250) — compile-verified
//
#include <hip/hip_runtime.h>
#include <hip/hip_bf16.h>

// ---------------------------------------------------------------------------
// MHA forward for gfx1250 (MI455X), all-f32 via V_WMMA_F32_16X16X4_F32.
// B=2, S=2048, D=1024, H=16, DK=64.  Reference bug preserved: V = K.
// d_out = [ output (B*S*D) | attn_weights.sum(heads) (B*S*S) ]  (f32)
// d_ws  = Q (B*S*D) | K (B*S*D) | concat (B*S*D)  -> 48 MB f32 scratch.
// ---------------------------------------------------------------------------

#define B_DIM 2
#define S_LEN 2048
#define D_DIM 1024
#define H_NUM 16
#define DK_DIM 64
#define SROW 2064              // padded LDS row stride (floats): 2064 % 64 == 16
#define NEG_BIG (-3.0e38f)

typedef __attribute__((ext_vector_type(2))) float v2f;
typedef __attribute__((ext_vector_type(8))) float v8f;

#define WMMA_F32(A, Bv, Cv) \
    __builtin_amdgcn_wmma_f32_16x16x4_f32(false, (A), false, (Bv), (short)0, \
                                          (Cv), false, false)

// y[m][n] = sum_k X[m][k] * W[n][k] + bias[n]    (torch Linear: X @ W.T + b)
// grid: (M/128, D/64); block: 256 threads = 8 waves.
// Each wave owns a 16(M) x 64(N) strip: one A load feeds 4 WMMAs (4 N-tiles),
// raising WMMA-per-load from 1:2 to 4:5.
__global__ __launch_bounds__(256) void proj_gemm(const float* __restrict__ X,
                                                 const float* __restrict__ W,
                                                 const float* __restrict__ bias,
                                                 float* __restrict__ Y) {
    const int tid  = threadIdx.x;
    const int wave = tid >> 5;
    const int lane = tid & 31;
    const int l15  = lane & 15;
    const int lhi  = lane >> 4;        // 0: K+0/1 half, 1: K+2/3 half
    const int ks   = lhi * 2;

    const int mbase = blockIdx.x * 128 + wave * 16;
    const int nbase = blockIdx.y * 64;

    const float* __restrict__ arow = X + (size_t)(mbase + l15) * D_DIM + ks;
    const float* __restrict__ brow = W + (size_t)(nbase + l15) * D_DIM + ks;

    v8f acc0 = {}, acc1 = {}, acc2 = {}, acc3 = {};
    for (int k = 0; k < D_DIM; k += 4) {
        v2f a;  a.x  = arow[k];                a.y  = arow[k + 1];
        v2f b0; b0.x = brow[k];                b0.y = brow[k + 1];
        v2f b1; b1.x = brow[k + 16 * D_DIM];   b1.y = brow[k + 16 * D_DIM + 1];
        v2f b2; b2.x = brow[k + 32 * D_DIM];   b2.y = brow[k + 32 * D_DIM + 1];
        v2f b3; b3.x = brow[k + 48 * D_DIM];   b3.y = brow[k + 48 * D_DIM + 1];
        acc0 = WMMA_F32(a, b0, acc0);
        acc1 = WMMA_F32(a, b1, acc1);
        acc2 = WMMA_F32(a, b2, acc2);
        acc3 = WMMA_F32(a, b3, acc3);
    }

    const int mrow0 = mbase + (lhi ? 8 : 0);
    v8f accs[4] = {acc0, acc1, acc2, acc3};
#pragma unroll
    for (int j = 0; j < 4; ++j) {
        const int nCol = nbase + j * 16 + l15;
        const float bv = bias[nCol];
#pragma unroll
        for (int v = 0; v < 8; ++v)
            Y[(size_t)(mrow0 + v) * D_DIM + nCol] = accs[j][v] + bv;
    }
}

// One block per (batch b, 16-query-row tile qt).  Loops all heads so the
// head-summed attention matrix is accumulated by a single deterministic owner.
// LDS: full score/prob rows for the 16 queries (16 x 2064 f32 ~129KB) +
//      4KB staging for the split-K P@V reduction.  Fits CDNA5 320KB WGP LDS.
__global__ __launch_bounds__(256) void attn_kernel(const float* __restrict__ Q,
                                                   const float* __restrict__ K,
                                                   float* __restrict__ attnsum,
                                                   float* __restrict__ C) {
    __shared__ float p[16 * SROW];     // scores -> probabilities
    __shared__ float red[4 * 16 * 16]; // split-K partials, 4 N-tiles

    const int qt = blockIdx.x;         // 0..127
    const int b  = blockIdx.y;         // 0..1
    const int q0 = qt * 16;

    const int tid  = threadIdx.x;
    const int wave = tid >> 5;
    const int lane = tid & 31;
    const int l15  = lane & 15;
    const int lhi  = lane >> 4;
    const int ks   = lhi * 2;
    const int rbase = lhi ? 8 : 0;

    const size_t bOff = (size_t)b * S_LEN * D_DIM;
    const int row = tid >> 4;          // softmax: 16 threads per query row
    const int c0  = tid & 15;

    for (int h = 0; h < H_NUM; ++h) {
        const float* __restrict__ Qh = Q + bOff + h * DK_DIM;
        const float* __restrict__ Kh = K + bOff + h * DK_DIM;

        // ---- 1. scores = (Q @ K^T) * DK^-0.5 into LDS, causal masked ----
        // Each wave handles PAIRS of adjacent 16-key tiles so the A (Q-tile)
        // load is shared by two WMMAs.  All branches are wave-uniform.
        const float* __restrict__ aq = Qh + (size_t)(q0 + l15) * D_DIM + ks;
        for (int pt = wave; pt < S_LEN / 32; pt += 8) {
            const int kt0  = pt * 2;
            const int col0 = kt0 * 16 + l15;
            const int col1 = col0 + 16;
            if (kt0 <= qt) {
                const float* __restrict__ bk =
                    Kh + (size_t)(kt0 * 16 + l15) * D_DIM + ks;
                if (kt0 + 1 <= qt) {               // both tiles live
                    v8f acc0 = {}, acc1 = {};
#pragma unroll
                    for (int k = 0; k < DK_DIM; k += 4) {
                        v2f a;  a.x  = aq[k];     a.y  = aq[k + 1];
                        v2f b0; b0.x = bk[k];     b0.y = bk[k + 1];
                        v2f b1; b1.x = bk[k + 16 * D_DIM];
                                b1.y = bk[k + 16 * D_DIM + 1];
                        acc0 = WMMA_F32(a, b0, acc0);
                        acc1 = WMMA_F32(a, b1, acc1);
                    }
#pragma unroll
                    for (int v = 0; v < 8; ++v) {
                        const int r = rbase + v;
                        float s0 = acc0[v] * 0.125f;          // DK^-0.5
                        float s1 = acc1[v] * 0.125f;
                        if (col0 > q0 + r) s0 = NEG_BIG;       // causal
                        if (col1 > q0 + r) s1 = NEG_BIG;
                        p[r * SROW + col0] = s0;
                        p[r * SROW + col1] = s1;
                    }
                } else {                            // diagonal pair: tile1 dead
                    v8f acc0 = {};
#pragma unroll
                    for (int k = 0; k < DK_DIM; k += 4) {
                        v2f a;  a.x  = aq[k]; a.y  = aq[k + 1];
                        v2f b0; b0.x = bk[k]; b0.y = bk[k + 1];
                        acc0 = WMMA_F32(a, b0, acc0);
                    }
#pragma unroll
                    for (int v = 0; v < 8; ++v) {
                        const int r = rbase + v;
                        float s0 = acc0[v] * 0.125f;
                        if (col0 > q0 + r) s0 = NEG_BIG;
                        p[r * SROW + col0] = s0;
                        p[r * SROW + col1] = NEG_BIG;
                    }
                }
            } else {                                // fully masked pair
#pragma unroll
                for (int v = 0; v < 8; ++v) {
                    p[(rbase + v) * SROW + col0] = NEG_BIG;
                    p[(rbase + v) * SROW + col1] = NEG_BIG;
                }
            }
        }
        __syncthreads();

        // ---- 2. softmax per query row (16 lanes per row) ----
        float* __restrict__ prow = p + row * SROW;
        float mx = NEG_BIG;
        for (int c = c0; c < S_LEN; c += 16) mx = fmaxf(mx, prow[c]);
#pragma unroll
        for (int m = 8; m; m >>= 1) mx = fmaxf(mx, __shfl_xor(mx, m, 16));
        float sum = 0.0f;
        for (int c = c0; c < S_LEN; c += 16) {
            float e = __expf(prow[c] - mx);
            prow[c] = e;
            sum += e;
        }
#pragma unroll
        for (int m = 8; m; m >>= 1) sum += __shfl_xor(sum, m, 16);
        const float inv = 1.0f / sum;
        for (int c = c0; c < S_LEN; c += 16) prow[c] *= inv;

        // ---- 3. accumulate head-sum of attention weights (owner block) ----
        float* __restrict__ as =
            attnsum + ((size_t)b * S_LEN + q0 + row) * S_LEN;
        if (h == 0) { for (int c = c0; c < S_LEN; c += 16) as[c]  = prow[c]; }
        else        { for (int c = c0; c < S_LEN; c += 16) as[c] += prow[c]; }
        __syncthreads();

        // ---- 4. atn = P @ V  (V == K by the reference bug) ----
        // waves 0..3 -> N-tile 0..3, K in [0,1024); waves 4..7 same N-tiles,
        // K in [1024,2048); partials reduced through LDS staging.  B loads
        // are coalesced 16-lane reads of one V row per half-wave.
        const int ntile = wave & 3;
        const int khalf = wave >> 2;
        const int nabs  = h * DK_DIM + ntile * 16;
        const float* __restrict__ Vh = K + bOff + nabs + l15;
        const int kstart = khalf * (S_LEN / 2);
        const int klim   = q0 + 16;                       // keys beyond are 0
        const int kend   = (kstart + S_LEN / 2 < klim) ? kstart + S_LEN / 2
                                                       : klim;
        v8f acc = {};
        for (int k = kstart; k < kend; k += 4) {          // wave-uniform
            const float* __restrict__ pr = p + l15 * SROW + k + ks;
            v2f a;  a.x  = pr[0]; a.y = pr[1];
            v2f bb;
            bb.x = Vh[(size_t)(k + ks)     * D_DIM];
            bb.y = Vh[(size_t)(k + ks + 1) * D_DIM];
            acc = WMMA_F32(a, bb, acc);
        }
        if (khalf == 1) {
#pragma unroll
            for (int v = 0; v < 8; ++v)
                red[(ntile * 16 + rbase + v) * 16 + l15] = acc[v];
        }
        __syncthreads();
        if (khalf == 0) {
#pragma unroll
            for (int v = 0; v < 8; ++v) {
                const int r = rbase + v;
                const float val = acc[v] + red[(ntile * 16 + r) * 16 + l15];
                C[((size_t)b * S_LEN + q0 + r) * D_DIM + nabs + l15] = val;
            }
        }
        __syncthreads();   // p and red are reused next head
    }
}

extern "C" void kernel_launch(void* const* d_in, const int* in_sizes, int n_in,
                              void* d_out, int out_size, void* d_ws, size_t ws_size,
                              hipStream_t stream) {
    const float* x    = (const float*)d_in[0];
    const float* z    = (const float*)d_in[1];
    // d_in[2] = causal mask (tril) -- applied analytically (col <= row)
    const float* wq_w = (const float*)d_in[3];
    const float* wq_b = (const float*)d_in[4];
    const float* wk_w = (const float*)d_in[5];
    const float* wk_b = (const float*)d_in[6];
    // wv_w / wv_b (d_in[7], d_in[8]) are dead params in the reference
    const float* wo_w = (const float*)d_in[9];
    const float* wo_b = (const float*)d_in[10];

    const size_t MD = (size_t)B_DIM * S_LEN * D_DIM;   // 4096*1024
    float* Qbuf = (float*)d_ws;
    float* Kbuf = Qbuf + MD;
    float* Cbuf = Kbuf + MD;

    float* out     = (float*)d_out;                    // [B,S,D]
    float* attnsum = out + MD;                         // [B,S,S]

    const dim3 gp(B_DIM * S_LEN / 128, D_DIM / 64);    // (32, 16)
    const dim3 blk(256);

    proj_gemm<<<gp, blk, 0, stream>>>(x, wq_w, wq_b, Qbuf);   // Q
    proj_gemm<<<gp, blk, 0, stream>>>(z, wk_w, wk_b, Kbuf);   // K (== V)

    attn_kernel<<<dim3(S_LEN / 16, B_DIM), blk, 0, stream>>>(Qbuf, Kbuf,
                                                             attnsum, Cbuf);

    proj_gemm<<<gp, blk, 0, stream>>>(Cbuf, wo_w, wo_b, out); // output proj
}